// DGCNN_37744172597611
// MI455X (gfx1250) — compile-verified
//
#include <hip/hip_runtime.h>

#define BATCH 8
#define NPTS  4096
#define KNN   20

typedef __attribute__((ext_vector_type(2))) float v2f;
typedef __attribute__((ext_vector_type(8))) float v8f;

__device__ __forceinline__ v8f wmma4(v2f a, v2f b, v8f c) {
  // V_WMMA_F32_16X16X4_F32: D(16x16,f32) = A(16x4,f32) x B(4x16,f32) + C
  return __builtin_amdgcn_wmma_f32_16x16x4_f32(
      /*neg_a=*/false, a, /*neg_b=*/false, b,
      /*c_mod=*/(short)0, c, /*reuse_a=*/false, /*reuse_b=*/false);
}

// ---------------------------------------------------------------------------
// Kernel 1: F = MLP1(x[batch 0])   (4096 x 64)
// (gather bug in reference: all batches gather batch-0 rows -> precompute once)
// ---------------------------------------------------------------------------
__global__ void mlp1_kernel(const float* __restrict__ x,
                            const float* __restrict__ w1, const float* __restrict__ b1,
                            const float* __restrict__ w2, const float* __restrict__ b2,
                            const float* __restrict__ w3, const float* __restrict__ b3,
                            float* __restrict__ F) {
  int p = blockIdx.x;        // point in batch 0
  int o = threadIdx.x;       // output channel (64 threads)
  __shared__ float s1[64], s2[64];
  float x0 = x[p * 3 + 0], x1 = x[p * 3 + 1], x2 = x[p * 3 + 2];
  float h = x0 * w1[o] + x1 * w1[64 + o] + x2 * w1[128 + o] + b1[o];
  s1[o] = fmaxf(h, 0.f);
  __syncthreads();
  float a = b2[o];
  #pragma unroll 8
  for (int i = 0; i < 64; ++i) a += s1[i] * w2[i * 64 + o];
  s2[o] = fmaxf(a, 0.f);
  __syncthreads();
  float c = b3[o];
  #pragma unroll 8
  for (int i = 0; i < 64; ++i) c += s2[i] * w3[i * 64 + o];
  F[p * 64 + o] = c;
}

// ---------------------------------------------------------------------------
// Kernel 2: per-batch KNN on xyz (C=3). One thread per query point.
// Top-20 kept in LDS with replace-min; k-order irrelevant (max-pooled later).
// ---------------------------------------------------------------------------
__global__ void knn_xyz_kernel(const float* __restrict__ x, int* __restrict__ idx1) {
  const int T = 256;
  int b   = blockIdx.x >> 4;                      // 16 blocks per batch
  int i   = ((blockIdx.x & 15) * T) + threadIdx.x;
  int tid = threadIdx.x;
  __shared__ float topv[KNN * T];
  __shared__ int   topi[KNN * T];
  const float* xb = x + (size_t)b * NPTS * 3;
  float qx = xb[i * 3 + 0], qy = xb[i * 3 + 1], qz = xb[i * 3 + 2];
  #pragma unroll
  for (int k = 0; k < KNN; ++k) topv[k * T + tid] = -3.0e38f;
  float minv = -3.0e38f; int minpos = 0;
  for (int j = 0; j < NPTS; ++j) {              // j uniform -> broadcast loads
    float dx = qx - xb[j * 3 + 0];
    float dy = qy - xb[j * 3 + 1];
    float dz = qz - xb[j * 3 + 2];
    float v = -(dx * dx + dy * dy + dz * dz);   // pd = -||d||^2 (maximize)
    if (v > minv) {
      topv[minpos * T + tid] = v; topi[minpos * T + tid] = j;
      float mv = 3.0e38f; int mp = 0;
      #pragma unroll
      for (int k = 0; k < KNN; ++k) {
        float tv = topv[k * T + tid];
        if (tv < mv) { mv = tv; mp = k; }
      }
      minv = mv; minpos = mp;
    }
  }
  int* dst = idx1 + ((size_t)b * NPTS + i) * KNN;
  #pragma unroll
  for (int k = 0; k < KNN; ++k) dst[k] = topi[k * T + tid];
}

// ---------------------------------------------------------------------------
// Kernel 3: h[q,:] = max_k F[idx1[q,k],:]  and  sqh[q] = ||h[q]||^2
// 4 queries per 256-thread block (64 channels each).
// ---------------------------------------------------------------------------
__global__ void gather_max64_kernel(const float* __restrict__ F, const int* __restrict__ idx1,
                                    float* __restrict__ h, float* __restrict__ sqh) {
  int tid = threadIdx.x;
  int q = blockIdx.x * 4 + (tid >> 6);
  int c = tid & 63;
  const int* ix = idx1 + (size_t)q * KNN;
  float v = -3.0e38f;
  #pragma unroll
  for (int k = 0; k < KNN; ++k) v = fmaxf(v, F[ix[k] * 64 + c]);
  h[(size_t)q * 64 + c] = v;
  __shared__ float red[256];
  red[tid] = v * v;
  __syncthreads();
  if (c == 0) {
    float s = 0.f;
    #pragma unroll
    for (int t = 0; t < 64; ++t) s += red[tid + t];
    sqh[q] = s;
  }
}

// ---------------------------------------------------------------------------
// Kernel 4: per-batch KNN on 64-d features via f32 WMMA (H x H^T).
// One wave per 16-row query tile. Software-pipelined: B fragments for tile
// t+1 are issued before tile t's WMMA chain consumes its (already in-flight)
// fragments, so the compiler waits with 16 loads still outstanding instead of
// stalling per-load. Dual accumulators break the D->C WMMA dependency chain.
// ---------------------------------------------------------------------------
__global__ void knn_feat_kernel(const float* __restrict__ h, const float* __restrict__ sqh,
                                int* __restrict__ idx2) {
  int b    = blockIdx.x >> 8;
  int i0   = (blockIdx.x & 255) * 16;
  int lane = threadIdx.x;
  int m = lane & 15, hi = lane >> 4;
  const float* Hb = h   + (size_t)b * NPTS * 64;
  const float* Sb = sqh + (size_t)b * NPTS;

  // A fragments: 16x64 query tile, K chunks of 4 (lane L: row m, k = 4c+2*hi)
  v2f a[16];
  {
    const float* arow = Hb + (size_t)(i0 + m) * 64 + 2 * hi;
    #pragma unroll
    for (int c = 0; c < 16; ++c) a[c] = *(const v2f*)(arow + 4 * c);
  }
  float sqi[8];
  #pragma unroll
  for (int r = 0; r < 8; ++r) sqi[r] = Sb[i0 + r + 8 * hi];

  __shared__ float tile[16 * 17];       // pad 17 -> conflict-free row reads
  __shared__ float topv[KNN * 16];
  __shared__ int   topi[KNN * 16];
  float minv = -3.0e38f; int minpos = 0;
  if (lane < 16) {
    #pragma unroll
    for (int k = 0; k < KNN; ++k) topv[k * 16 + lane] = -3.0e38f;
  }
  __syncthreads();

  // B fragments: B[k][n] = H[jt*16+n][k]; same float2 pattern as A (symmetry)
  const float* bbase = Hb + (size_t)m * 64 + 2 * hi;

  v2f b0[16], b1[16];
  #pragma unroll
  for (int c = 0; c < 16; ++c) b0[c] = *(const v2f*)(bbase + 4 * c);

  // Per-tile compute: WMMA chain (even/odd accumulators) + pd + top-20 merge.
  auto compute = [&](int jt, const v2f* bf) {
    v8f accE = {0.f,0.f,0.f,0.f,0.f,0.f,0.f,0.f};
    v8f accO = {0.f,0.f,0.f,0.f,0.f,0.f,0.f,0.f};
    #pragma unroll
    for (int c = 0; c < 8; ++c) {
      accE = wmma4(a[2*c],     bf[2*c],     accE);
      accO = wmma4(a[2*c + 1], bf[2*c + 1], accO);
    }
    float sqj = Sb[jt * 16 + m];        // lane's column n = lane&15
    #pragma unroll
    for (int r = 0; r < 8; ++r)
      tile[(r + 8*hi)*17 + m] = 2.f*(accE[r] + accO[r]) - sqi[r] - sqj;
    __syncthreads();
    if (lane < 16) {                    // lane owns query row `lane`
      #pragma unroll
      for (int n = 0; n < 16; ++n) {
        float v = tile[lane * 17 + n];
        if (v > minv) {
          topv[minpos * 16 + lane] = v;
          topi[minpos * 16 + lane] = jt * 16 + n;
          float mv = 3.0e38f; int mp = 0;
          #pragma unroll
          for (int k = 0; k < KNN; ++k) {
            float tv = topv[k * 16 + lane];
            if (tv < mv) { mv = tv; mp = k; }
          }
          minv = mv; minpos = mp;
        }
      }
    }
    __syncthreads();
  };

  for (int jt = 0; jt < NPTS / 16; jt += 2) {
    {                                   // issue tile jt+1 loads (no wait)
      const float* p = bbase + (size_t)(jt + 1) * 16 * 64;
      #pragma unroll
      for (int c = 0; c < 16; ++c) b1[c] = *(const v2f*)(p + 4 * c);
    }
    compute(jt, b0);
    {                                   // issue tile jt+2 loads (wraps on last)
      const float* p = bbase + (size_t)((jt + 2) & 255) * 16 * 64;
      #pragma unroll
      for (int c = 0; c < 16; ++c) b0[c] = *(const v2f*)(p + 4 * c);
    }
    compute(jt + 1, b1);
  }

  if (lane < 16) {
    int* dst = idx2 + ((size_t)b * NPTS + i0 + lane) * KNN;
    #pragma unroll
    for (int k = 0; k < KNN; ++k) dst[k] = topi[k * 16 + lane];
  }
}

// ---------------------------------------------------------------------------
// Kernel 5: g = h[batch0] @ w4 + b4   (4096 x 128)
// ---------------------------------------------------------------------------
__global__ void gemm_g_kernel(const float* __restrict__ h, const float* __restrict__ w4,
                              const float* __restrict__ b4, float* __restrict__ g) {
  int p = blockIdx.x, o = threadIdx.x;   // 128 threads
  __shared__ float s[64];
  if (o < 64) s[o] = h[(size_t)p * 64 + o];
  __syncthreads();
  float acc = b4[o];
  #pragma unroll 8
  for (int i = 0; i < 64; ++i) acc += s[i] * w4[i * 128 + o];
  g[(size_t)p * 128 + o] = acc;
}

// ---------------------------------------------------------------------------
// Kernel 6: h2[q,:] = max_k g[idx2[q,k],:]   (2 queries per 256-thread block)
// ---------------------------------------------------------------------------
__global__ void gather_max128_kernel(const float* __restrict__ g, const int* __restrict__ idx2,
                                     float* __restrict__ h2) {
  int tid = threadIdx.x;
  int q = blockIdx.x * 2 + (tid >> 7);
  int c = tid & 127;
  const int* ix = idx2 + (size_t)q * KNN;
  float v = -3.0e38f;
  #pragma unroll
  for (int k = 0; k < KNN; ++k) v = fmaxf(v, g[ix[k] * 128 + c]);
  h2[(size_t)q * 128 + c] = v;
}

// ---------------------------------------------------------------------------
// Kernel 7: out = h2 @ w5 + b5  ((32768x128)@(128x1024)) via f32 WMMA.
// One wave per (16-row M tile, 64-col N group); A reused across 4 N tiles.
// All 64 B scalars for an N tile are preloaded (one batched wait, loads are
// 64B-contiguous per half-wave); dual accumulators for WMMA ILP.
// ---------------------------------------------------------------------------
__global__ void final_gemm_kernel(const float* __restrict__ h2, const float* __restrict__ w5,
                                  const float* __restrict__ b5, float* __restrict__ out) {
  int mt   = blockIdx.x >> 4;
  int ng   = blockIdx.x & 15;
  int lane = threadIdx.x;
  int m = lane & 15, hi = lane >> 4;
  const float* arow = h2 + (size_t)(mt * 16 + m) * 128 + 2 * hi;
  v2f a[32];
  #pragma unroll
  for (int c = 0; c < 32; ++c) a[c] = *(const v2f*)(arow + 4 * c);
  for (int nt = 0; nt < 4; ++nt) {
    int n = ng * 64 + nt * 16 + m;      // lane's output column
    float bx[32], by[32];
    #pragma unroll
    for (int c = 0; c < 32; ++c) {      // batched issue: one wait covers all
      bx[c] = w5[(size_t)(4 * c + 2 * hi) * 1024 + n];
      by[c] = w5[(size_t)(4 * c + 2 * hi + 1) * 1024 + n];
    }
    float bias = b5[n];
    v8f accE, accO;
    #pragma unroll
    for (int r = 0; r < 8; ++r) { accE[r] = bias; accO[r] = 0.f; }
    #pragma unroll
    for (int c = 0; c < 16; ++c) {
      v2f bb0; bb0.x = bx[2*c];     bb0.y = by[2*c];
      accE = wmma4(a[2*c], bb0, accE);
      v2f bb1; bb1.x = bx[2*c + 1]; bb1.y = by[2*c + 1];
      accO = wmma4(a[2*c + 1], bb1, accO);
    }
    float* orow = out + (size_t)mt * 16 * 1024 + n;
    #pragma unroll
    for (int r = 0; r < 8; ++r) orow[(size_t)(r + 8 * hi) * 1024] = accE[r] + accO[r];
  }
}

// ---------------------------------------------------------------------------
extern "C" void kernel_launch(void* const* d_in, const int* in_sizes, int n_in,
                              void* d_out, int out_size, void* d_ws, size_t ws_size,
                              hipStream_t stream) {
  const float* x  = (const float*)d_in[0];
  const float* w1 = (const float*)d_in[1];
  const float* b1 = (const float*)d_in[2];
  const float* w2 = (const float*)d_in[3];
  const float* b2 = (const float*)d_in[4];
  const float* w3 = (const float*)d_in[5];
  const float* b3 = (const float*)d_in[6];
  const float* w4 = (const float*)d_in[7];
  const float* b4 = (const float*)d_in[8];
  const float* w5 = (const float*)d_in[9];
  const float* b5 = (const float*)d_in[10];
  float* out = (float*)d_out;

  // Workspace layout (floats), ~34 MB total
  float* F    = (float*)d_ws;                      // 4096*64
  float* h    = F   + (size_t)NPTS * 64;           // 8*4096*64
  float* sqh  = h   + (size_t)BATCH * NPTS * 64;   // 8*4096
  float* g    = sqh + (size_t)BATCH * NPTS;        // 4096*128
  float* h2   = g   + (size_t)NPTS * 128;          // 8*4096*128
  int*   idx1 = (int*)(h2 + (size_t)BATCH * NPTS * 128);  // 8*4096*20
  int*   idx2 = idx1 + (size_t)BATCH * NPTS * KNN;        // 8*4096*20

  mlp1_kernel<<<NPTS, 64, 0, stream>>>(x, w1, b1, w2, b2, w3, b3, F);
  knn_xyz_kernel<<<BATCH * 16, 256, 0, stream>>>(x, idx1);
  gather_max64_kernel<<<(BATCH * NPTS) / 4, 256, 0, stream>>>(F, idx1, h, sqh);
  knn_feat_kernel<<<BATCH * (NPTS / 16), 32, 0, stream>>>(h, sqh, idx2);
  gemm_g_kernel<<<NPTS, 128, 0, stream>>>(h, w4, b4, g);
  gather_max128_kernel<<<(BATCH * NPTS) / 2, 256, 0, stream>>>(g, idx2, h2);
  final_gemm_kernel<<<(BATCH * NPTS / 16) * 16, 32, 0, stream>>>(h2, w5, b5, out);
}